// GraphConvLayer_45612552684102
// MI455X (gfx1250) — compile-verified
//
#include <hip/hip_runtime.h>

typedef __attribute__((ext_vector_type(2))) float v2f;
typedef __attribute__((ext_vector_type(8))) float v8f;

#define D 64  // D_IN == D_OUT == 64

// ---------------------------------------------------------------------------
// Kernel 1: zero the output accumulator, init degrees to 1.0 (self edge).
// ---------------------------------------------------------------------------
__global__ void gcn_init_kernel(float* __restrict__ out,
                                float* __restrict__ sdeg,
                                float* __restrict__ rdeg, int n) {
    long long i = (long long)blockIdx.x * blockDim.x + threadIdx.x;
    if (i < n) { sdeg[i] = 1.0f; rdeg[i] = 1.0f; }
    if (i < (long long)n * D) out[i] = 0.0f;
}

// ---------------------------------------------------------------------------
// Kernel 2: degree accumulation over edges (float atomics are exact for counts).
// ---------------------------------------------------------------------------
__global__ void gcn_degree_kernel(const int* __restrict__ senders,
                                  const int* __restrict__ receivers,
                                  float* __restrict__ sdeg,
                                  float* __restrict__ rdeg, int e) {
    int i = blockIdx.x * blockDim.x + threadIdx.x;
    if (i < e) {
        atomicAdd(&sdeg[senders[i]], 1.0f);
        atomicAdd(&rdeg[receivers[i]], 1.0f);
    }
}

// ---------------------------------------------------------------------------
// Kernel 3: h = (x @ W^T + bias) * rsqrt(max(sender_deg,1))
// One wave32 per 16x16 output tile using V_WMMA_F32_16X16X4_F32 (fp32 exact).
// Block = 128 threads = 4 waves; wave w handles column tile w (4 tiles of 16
// cover D_OUT=64); blockIdx.x handles the row tile.
//
// A (16x4 f32, 2 VGPRs): lane 0-15 -> M=lane, K={0,1}; lane 16-31 -> K={2,3}.
// B loaded column-major: lane selects N, VGPR+lane-half select K, so
// b[v] = W[n][k] (which is (W^T)[k][n]).
// C/D (16x16 f32, 8 VGPRs): lane 0-15 -> N=lane, M=v; lane 16-31 -> M=8+v.
// ---------------------------------------------------------------------------
__global__ __launch_bounds__(128) void gcn_gemm_wmma_kernel(
    const float* __restrict__ x, const float* __restrict__ weight,
    const float* __restrict__ bias, const float* __restrict__ sdeg,
    float* __restrict__ h, int n) {
    const int wave = threadIdx.x >> 5;   // 0..3 -> column tile
    const int lane = threadIdx.x & 31;
    const int lo   = lane & 15;
    const int hi   = lane >> 4;          // K-group select
    const int row0 = blockIdx.x * 16;
    const int col0 = wave * 16;
    if (row0 >= n) return;

    const int m_a = (row0 + lo < n) ? (row0 + lo) : (n - 1);  // A row (clamped)
    const float* __restrict__ xr = x + (long long)m_a * D;
    const float* __restrict__ wr = weight + (long long)(col0 + lo) * D;

    const float bval = bias[col0 + lo];  // C element's N index == lo
    v8f c;
    #pragma unroll
    for (int v = 0; v < 8; ++v) c[v] = bval;

    #pragma unroll
    for (int k0 = 0; k0 < D; k0 += 4) {
        const int k = k0 + 2 * hi;
        v2f a, b;
        a.x = xr[k];     a.y = xr[k + 1];
        b.x = wr[k];     b.y = wr[k + 1];
        c = __builtin_amdgcn_wmma_f32_16x16x4_f32(
                /*neg_a=*/false, a, /*neg_b=*/false, b,
                /*c_mod=*/(short)0, c, /*reuse_a=*/false, /*reuse_b=*/false);
    }

    #pragma unroll
    for (int v = 0; v < 8; ++v) {
        const int m = row0 + v + 8 * hi;
        if (m < n) {
            const float s = __frsqrt_rn(fmaxf(sdeg[m], 1.0f));
            h[(long long)m * D + col0 + lo] = c[v] * s;
        }
    }
}

// ---------------------------------------------------------------------------
// Kernel 4: edge scatter-add. One thread per (edge, feature):
// out[receivers[e]][f] += h[senders[e]][f]. Lane-uniform index loads coalesce;
// row reads are 128B-contiguous per wave.
// ---------------------------------------------------------------------------
__global__ void gcn_scatter_kernel(const int* __restrict__ senders,
                                   const int* __restrict__ receivers,
                                   const float* __restrict__ h,
                                   float* __restrict__ out, int e) {
    long long gid = (long long)blockIdx.x * blockDim.x + threadIdx.x;
    int edge = (int)(gid >> 6);
    int f    = (int)(gid & (D - 1));
    if (edge < e) {
        int s = senders[edge];
        int r = receivers[edge];
        atomicAdd(&out[(long long)r * D + f], h[(long long)s * D + f]);
    }
}

// ---------------------------------------------------------------------------
// Kernel 5: add self-loop term, receiver-side normalization, leaky ReLU.
// ---------------------------------------------------------------------------
__global__ void gcn_finalize_kernel(const float* __restrict__ h,
                                    const float* __restrict__ rdeg,
                                    float* __restrict__ out, long long total) {
    long long i = (long long)blockIdx.x * blockDim.x + threadIdx.x;
    if (i < total) {
        int row = (int)(i >> 6);
        float s = __frsqrt_rn(fmaxf(rdeg[row], 1.0f));
        float v = (out[i] + h[i]) * s;      // h[i] = self-edge contribution
        out[i] = (v >= 0.0f) ? v : 0.01f * v;
    }
}

// ---------------------------------------------------------------------------
extern "C" void kernel_launch(void* const* d_in, const int* in_sizes, int n_in,
                              void* d_out, int out_size, void* d_ws, size_t ws_size,
                              hipStream_t stream) {
    const float* x         = (const float*)d_in[0];
    const int*   senders   = (const int*)  d_in[1];
    const int*   receivers = (const int*)  d_in[2];
    const float* weight    = (const float*)d_in[3];
    const float* bias      = (const float*)d_in[4];
    float*       out       = (float*)d_out;

    const int n = in_sizes[0] / D;   // 50000
    const int e = in_sizes[1];       // 800000

    // Workspace layout: h [n*64], sender_deg [n], receiver_deg [n]
    float* h    = (float*)d_ws;
    float* sdeg = h + (size_t)n * D;
    float* rdeg = sdeg + n;

    const long long total = (long long)n * D;

    gcn_init_kernel<<<(int)((total + 255) / 256), 256, 0, stream>>>(out, sdeg, rdeg, n);
    gcn_degree_kernel<<<(e + 255) / 256, 256, 0, stream>>>(senders, receivers, sdeg, rdeg, e);
    gcn_gemm_wmma_kernel<<<(n + 15) / 16, 128, 0, stream>>>(x, weight, bias, sdeg, h, n);

    const long long sc_threads = (long long)e * D;
    gcn_scatter_kernel<<<(int)((sc_threads + 255) / 256), 256, 0, stream>>>(senders, receivers, h, out, e);
    gcn_finalize_kernel<<<(int)((total + 255) / 256), 256, 0, stream>>>(h, rdeg, out, total);
}